// Correlation_72164040507827
// MI455X (gfx1250) — compile-verified
//
#include <hip/hip_runtime.h>

typedef __attribute__((ext_vector_type(2))) float    v2f;
typedef __attribute__((ext_vector_type(8))) float    v8f;
typedef __attribute__((ext_vector_type(4))) unsigned v4u;
typedef __attribute__((ext_vector_type(8))) unsigned v8u;

#define BATCH 8
#define CH    256
#define HH    96
#define WW    128
#define MD    4
#define ND    9                 // displacements per axis (2*MD+1)
#define CC    4                 // channels per LDS chunk (K granule fits one WMMA step x1)
#define WP    160               // padded in2 row pitch in LDS (128 data + 32 pad dwords)
#define NCHUNK (CH / CC)        // 64

// LDS (double buffered, TDM-fed):
//   lds1[2][CC*WW]            : in1 chunk, plain [c][x], 2*512 floats   =  4 KB
//   lds2[2][16 + ND*CC*WP]    : 9 dy rows x CC ch, pitch WP, 16-float
//                               head halo; pad/halo regions are zeroed
//                               once and never written by TDM           = ~46 KB
#define L2SZ (16 + ND * CC * WP)   // 5776 floats per buffer

// ---- Tensor Data Mover: issue one 2D tile load (CC lines of WW floats) ----
// D# group0: count=1 | lds_addr | global_addr(57b) | type=2
// D# group1: data_size=4B, optional LDS padding (32 dwords after every 128),
//            tensor_dim0=WW, tensor_dim1=CH, tile_dim0 (0 => NOP),
//            tile_dim1=CC lines, line stride = H*W elements.
__device__ __forceinline__ void tdm_load_tile(unsigned lds_byte_addr,
                                              const float* gptr,
                                              unsigned g1w0,
                                              unsigned tile_dim0)
{
    unsigned long long ga = (unsigned long long)(uintptr_t)gptr;
    v4u g0;
    g0[0] = 1u;                                          // count=1, user D#
    g0[1] = lds_byte_addr;                               // lds_addr [63:32]
    g0[2] = (unsigned)ga;                                // global_addr lo
    g0[3] = (unsigned)((ga >> 32) & 0x1FFFFFFu)          // global_addr [56:32]
          | (2u << 30);                                  // type=2 (image)
    v8u g1;
    g1[0] = g1w0;                                        // data_size / padding
    g1[1] = (unsigned)WW << 16;                          // tensor_dim0 = 128
    g1[2] = (unsigned)CH << 16;                          // tensor_dim1 = 256
    g1[3] = tile_dim0 << 16;                             // tile_dim0 (0 => NOP)
    g1[4] = (unsigned)CC;                                // tile_dim1 = CC, tile_dim2 = 0
    g1[5] = (unsigned)(HH * WW);                         // tensor_dim0_stride = 12288
    g1[6] = 0u;
    g1[7] = 0u;
    asm volatile("tensor_load_to_lds %0, %1" :: "s"(g0), "s"(g1) : "memory");
}

#define G1_PLAIN  (2u << 16)                                      // 4-byte elems
#define G1_PADDED ((2u << 16) | (1u << 20) | (6u << 22) | (31u << 25))
// pad_enable, pad_interval=6 (every 128 dwords), pad_amount=31 (32 dwords)

__global__ __launch_bounds__(256)
void corr_wmma_kernel(const float* __restrict__ in1,
                      const float* __restrict__ in2,
                      float* __restrict__ out)
{
    __shared__ float lds1[2][CC * WW];
    __shared__ float lds2[2][L2SZ];

    const int tid = threadIdx.x;
    const int b   = blockIdx.x / HH;
    const int y   = blockIdx.x % HH;

    // Zero both in2 buffers once: TDM never writes halo/pad regions, so the
    // zero-padded correlation borders (x and y) stay correct for all chunks.
    for (int i = tid; i < 2 * L2SZ; i += 256) ((float*)lds2)[i] = 0.f;

    const int lane   = tid & 31;
    const int wave   = tid >> 5;
    const int hl     = lane >> 4;   // 0: lanes 0-15 (K0/K1), 1: lanes 16-31 (K2/K3)
    const int m      = lane & 15;   // M (x1) for A, N (x2) for B/C/D
    const int x1base = wave * 16;   // 8 waves cover WW = 128

    const unsigned lds1_addr = (unsigned)(uintptr_t)&lds1[0][0];
    const unsigned lds2_addr = (unsigned)(uintptr_t)&lds2[0][0];

    v8f acc[ND][2];
    #pragma unroll
    for (int d = 0; d < ND; ++d)
        #pragma unroll
        for (int w = 0; w < 2; ++w)
            acc[d][w] = (v8f)0.f;

    const float* in1_b = in1 + (size_t)b * CH * HH * WW;
    const float* in2_b = in2 + (size_t)b * CH * HH * WW;

    __syncthreads();   // zeros visible before any TDM write

    // ---- prologue: stage chunk 0 into buffer 0 (wave 0 drives the TDM) ----
    if (wave == 0) {
        tdm_load_tile(lds1_addr, in1_b + ((size_t)0 * HH + y) * WW, G1_PLAIN, WW);
        #pragma unroll
        for (int dyi = 0; dyi < ND; ++dyi) {
            int row = y + dyi - MD;
            int ok  = (row >= 0 && row < HH);
            int rc  = ok ? row : 0;
            tdm_load_tile(lds2_addr + (16 + dyi * CC * WP) * 4,
                          in2_b + ((size_t)0 * HH + rc) * WW,
                          G1_PADDED, ok ? WW : 0);   // tile_dim0 = 0 => NOP
        }
    }

    for (int ci = 0; ci < NCHUNK; ++ci) {
        const int buf = ci & 1;

        if (wave == 0) {
            if (ci + 1 < NCHUNK) {
                // ---- stage chunk ci+1 into the other buffer, async ----
                const int   nb = buf ^ 1;
                const int   c1 = (ci + 1) * CC;
                tdm_load_tile(lds1_addr + nb * (CC * WW * 4),
                              in1_b + ((size_t)c1 * HH + y) * WW, G1_PLAIN, WW);
                #pragma unroll
                for (int dyi = 0; dyi < ND; ++dyi) {
                    int row = y + dyi - MD;
                    int ok  = (row >= 0 && row < HH);
                    int rc  = ok ? row : 0;
                    tdm_load_tile(lds2_addr + nb * (L2SZ * 4) + (16 + dyi * CC * WP) * 4,
                                  in2_b + ((size_t)c1 * HH + rc) * WW,
                                  G1_PADDED, ok ? WW : 0);
                }
                // 10 TDMs just issued for ci+1; tensor ops complete in order,
                // so cnt<=10 guarantees chunk ci is fully in LDS.
                __builtin_amdgcn_s_wait_tensorcnt(10);
            } else {
                __builtin_amdgcn_s_wait_tensorcnt(0);
            }
        }
        __syncthreads();   // publish chunk ci to all waves

        // ---- WMMA: banded Gram over this K=CC slice; A reused 18x ----
        const float* l1 = lds1[buf];
        const float* l2 = lds2[buf];
        {
            v2f a;
            a.x = l1[(hl * 2 + 0) * WW + x1base + m];
            a.y = l1[(hl * 2 + 1) * WW + x1base + m];   // -> ds_load_2addr_b32
            #pragma unroll
            for (int dyi = 0; dyi < ND; ++dyi) {
                #pragma unroll
                for (int w = 0; w < 2; ++w) {
                    int xi = 16 + (dyi * CC + hl * 2) * WP
                           + (x1base - MD + w * 16 + m);
                    v2f bf;
                    bf.x = l2[xi];
                    bf.y = l2[xi + WP];                 // -> ds_load_2addr_b32
                    acc[dyi][w] = __builtin_amdgcn_wmma_f32_16x16x4_f32(
                        false, a, false, bf, (short)0, acc[dyi][w], false, false);
                }
            }
        }
        __syncthreads();   // all reads done before buf is overwritten next iter
    }

    // ---- extract the +/-4 band, normalize (mean over C), store ----
    const float scale = 1.0f / (float)CH;
    #pragma unroll
    for (int dyi = 0; dyi < ND; ++dyi) {
        #pragma unroll
        for (int w = 0; w < 2; ++w) {
            #pragma unroll
            for (int r = 0; r < 8; ++r) {
                int x1 = x1base + r + 8 * hl;                 // C/D: VGPR r -> M=r(+8)
                int dx = (-MD + w * 16 + m) - (r + 8 * hl);   // x2 - x1
                if (dx >= -MD && dx <= MD) {
                    int d = dyi * ND + (dx + MD);             // (dy+4)*9 + (dx+4)
                    out[(((size_t)b * (ND * ND) + d) * HH + y) * WW + x1] =
                        acc[dyi][w][r] * scale;
                }
            }
        }
    }
}

extern "C" void kernel_launch(void* const* d_in, const int* in_sizes, int n_in,
                              void* d_out, int out_size, void* d_ws, size_t ws_size,
                              hipStream_t stream)
{
    (void)in_sizes; (void)n_in; (void)d_ws; (void)ws_size; (void)out_size;
    const float* in1 = (const float*)d_in[0];
    const float* in2 = (const float*)d_in[1];
    float* out = (float*)d_out;

    dim3 grid(BATCH * HH);   // 768 workgroups: one per (b, y) row
    dim3 block(256);         // 8 waves; wave w owns x1 tile [16w, 16w+15]
    corr_wmma_kernel<<<grid, block, 0, stream>>>(in1, in2, out);
}